// MultiVectorCELoss_35150012350881
// MI455X (gfx1250) — compile-verified
//
#include <hip/hip_runtime.h>
#include <cstdint>

// MultiVectorCE loss for MI455X (gfx1250), compile-only tuned.
// HBM-bound: 524MB traffic, floor ~22.5us @ 23.3 TB/s. Strategy:
//  - one wave32 per row, async global->LDS staging (ASYNCcnt path, NT hint:
//    data is streamed exactly once),
//  - 3 LDS passes (max; neg sums; positive losses), branchless selects so the
//    transcendental bodies never diverge, wave-shuffle reductions,
//  - deterministic final tree reduction (no float atomics).

#define C_DIM   1000
#define ROW_BYTES (C_DIM * 4)        // 4000, 16B aligned
#define SLOTS_PER_WAVE 1024          // padded row length (32 lanes x 8 float4)

__device__ __forceinline__ uint32_t lds_off(const void* p) {
  // addrspacecast(3->flat) puts the LDS byte offset in the low 32 bits.
  return (uint32_t)(uintptr_t)p;
}

__device__ __forceinline__ float wave_sum(float v) {
#pragma unroll
  for (int m = 16; m >= 1; m >>= 1) v += __shfl_xor(v, m, 32);
  return v;
}

__device__ __forceinline__ float wave_max(float v) {
#pragma unroll
  for (int m = 16; m >= 1; m >>= 1) v = fmaxf(v, __shfl_xor(v, m, 32));
  return v;
}

__global__ __launch_bounds__(256) void mvce_row_kernel(
    const float* __restrict__ outp, const float* __restrict__ tgtp,
    float* __restrict__ row_loss, int B) {
  // Per wave: 1024 floats of output + 1024 floats of target = 8KB.
  __shared__ float smem[8 * 2 * SLOTS_PER_WAVE];

  const int wave = threadIdx.x >> 5;
  const int lane = threadIdx.x & 31;
  const int row  = blockIdx.x * 8 + wave;
  if (row >= B) return;

  float* so = smem + wave * (2 * SLOTS_PER_WAVE);
  float* st = so + SLOTS_PER_WAVE;

  const uint64_t obase   = (uint64_t)(uintptr_t)outp;
  const uint64_t tbase   = (uint64_t)(uintptr_t)tgtp;
  const uint32_t rowbyte = (uint32_t)row * (uint32_t)ROW_BYTES;

  // ---- Stage row tile into LDS via async tensor-path loads (ASYNCcnt) ----
#pragma unroll
  for (int j = 0; j < 8; ++j) {
    const int f = j * 32 + lane;            // float4 slot index, 0..255
    if (f < C_DIM / 4) {                    // f < 250: real data
      const uint32_t voff = rowbyte + (uint32_t)f * 16u;
      const uint32_t lo   = lds_off(so + f * 4);
      const uint32_t lt   = lds_off(st + f * 4);
      asm volatile("global_load_async_to_lds_b128 %0, %1, %2 th:TH_LOAD_NT"
                   :: "v"(lo), "v"(voff), "s"(obase) : "memory");
      asm volatile("global_load_async_to_lds_b128 %0, %1, %2 th:TH_LOAD_NT"
                   :: "v"(lt), "v"(voff), "s"(tbase) : "memory");
    } else {                                // tail pad: inert sentinel
#pragma unroll
      for (int k = 0; k < 4; ++k) {
        so[f * 4 + k] = -1.0e30f;           // exp(o-m) -> 0, never wins max
        st[f * 4 + k] = 0.0f;               // negative with zero weight
      }
    }
  }
  // Wave-private tile: only need our own async writes visible.
  asm volatile("s_wait_asynccnt 0x0" ::: "memory");

  // ---- Pass 1: row max (stability shift) ----
  float m = -1.0e30f;
#pragma unroll
  for (int j = 0; j < 8; ++j) {
    const int f = j * 32 + lane;
    const float4 o4 = *reinterpret_cast<const float4*>(so + f * 4);
    m = fmaxf(m, fmaxf(fmaxf(o4.x, o4.y), fmaxf(o4.z, o4.w)));
  }
  m = wave_max(m);

  // ---- Pass 2: negative-set statistics Sneg, Tneg, Aneg (branchless) ----
  float S = 0.0f, T = 0.0f, A = 0.0f;
#pragma unroll
  for (int j = 0; j < 8; ++j) {
    const int f = j * 32 + lane;
    const float4 o4 = *reinterpret_cast<const float4*>(so + f * 4);
    const float4 t4 = *reinterpret_cast<const float4*>(st + f * 4);
    const float o[4] = {o4.x, o4.y, o4.z, o4.w};
    const float t[4] = {t4.x, t4.y, t4.z, t4.w};
#pragma unroll
    for (int k = 0; k < 4; ++k) {
      const float e   = expf(o[k] - m);     // sentinel slots underflow to 0
      const bool  neg = (t[k] <= 0.5f);
      S += neg ? e : 0.0f;
      T += neg ? t[k] : 0.0f;
      A += neg ? t[k] * o[k] : 0.0f;
    }
  }
  S = wave_sum(S);
  T = wave_sum(T);
  A = wave_sum(A);

  // ---- Pass 3: per-positive losses (branchless) ----
  float L = 0.0f, np = 0.0f;
#pragma unroll
  for (int j = 0; j < 8; ++j) {
    const int f = j * 32 + lane;
    const float4 o4 = *reinterpret_cast<const float4*>(so + f * 4);
    const float4 t4 = *reinterpret_cast<const float4*>(st + f * 4);
    const float o[4] = {o4.x, o4.y, o4.z, o4.w};
    const float t[4] = {t4.x, t4.y, t4.z, t4.w};
#pragma unroll
    for (int k = 0; k < 4; ++k) {
      const float e   = expf(o[k] - m);
      const float lse = m + logf(S + e);    // finite for sentinel slots too
      const float lp  = (T + t[k]) * lse - A - t[k] * o[k];
      const bool  pos = (t[k] > 0.5f);
      L  += pos ? lp : 0.0f;
      np += pos ? 1.0f : 0.0f;
    }
  }
  L  = wave_sum(L);
  np = wave_sum(np);

  if (lane == 0) {
    const float fallback = T * (m + logf(S)) - A;
    row_loss[row] = (np > 0.0f) ? (L / np) : fallback;
  }
}

__global__ __launch_bounds__(256) void mvce_reduce_kernel(
    const float* __restrict__ row_loss, float* __restrict__ out, int B) {
  __shared__ float red[256];
  float s = 0.0f;
  for (int i = threadIdx.x; i < B; i += 256) s += row_loss[i];
  red[threadIdx.x] = s;
  __syncthreads();
#pragma unroll
  for (int k = 128; k > 0; k >>= 1) {
    if ((int)threadIdx.x < k) red[threadIdx.x] += red[threadIdx.x + k];
    __syncthreads();
  }
  if (threadIdx.x == 0) out[0] = red[0] * (1.0f / (float)B);
}

extern "C" void kernel_launch(void* const* d_in, const int* in_sizes, int n_in,
                              void* d_out, int out_size, void* d_ws, size_t ws_size,
                              hipStream_t stream) {
  (void)n_in; (void)out_size; (void)ws_size;
  const float* outp = (const float*)d_in[0];
  const float* tgtp = (const float*)d_in[1];
  const int B = in_sizes[0] / C_DIM;      // 65536
  float* rl = (float*)d_ws;               // B floats of scratch (256 KB)

  const int grid = (B + 7) / 8;           // 8 rows (waves) per 256-thread block
  hipLaunchKernelGGL(mvce_row_kernel, dim3(grid), dim3(256), 0, stream,
                     outp, tgtp, rl, B);
  hipLaunchKernelGGL(mvce_reduce_kernel, dim3(1), dim3(256), 0, stream,
                     rl, (float*)d_out, B);
}